// DurationControl_78683800863384
// MI455X (gfx1250) — compile-verified
//
#include <hip/hip_runtime.h>

#define B_    16
#define C_    256
#define T_    2048
#define F_    256
#define GIN_  256
#define MELC_ 80
#define UMAX_ 1024
#define OUTC_ 592   // 2*C + MELC

typedef float v2f __attribute__((ext_vector_type(2)));
typedef float v8f __attribute__((ext_vector_type(8)));

// ---------------- 1. run-length dedup: segments, starts, mask ----------------
__global__ __launch_bounds__(16) void dedup_kernel(const int* __restrict__ ph,
                                                   int* __restrict__ dur,
                                                   int* __restrict__ start,
                                                   float* __restrict__ mask) {
  int b = threadIdx.x;
  if (b >= B_) return;
  const int* p = ph + (size_t)b * T_;
  int*   db = dur   + (size_t)b * UMAX_;
  int*   sb = start + (size_t)b * UMAX_;
  float* mb = mask  + (size_t)b * UMAX_;
  for (int i = 0; i < UMAX_; ++i) { db[i] = 0; sb[i] = 0; mb[i] = 0.f; }
  int u = -1, prev = -2147483647;
  for (int t = 0; t < T_; ++t) {
    int v = p[t];
    if (t == 0 || v != prev) {
      ++u;
      if (u < UMAX_) { sb[u] = t; mb[u] = 1.f; }
    }
    if (u >= 0 && u < UMAX_) db[u] += 1;
    prev = v;
  }
}

// ---------------- 2. mean-pool frames -> units (contiguous ranges) ----------
__global__ __launch_bounds__(256) void pool_kernel(const float* __restrict__ ec,
                                                   const float* __restrict__ ep,
                                                   const int* __restrict__ dur,
                                                   const int* __restrict__ start,
                                                   float* __restrict__ u_c,
                                                   float* __restrict__ u_p) {
  int u = blockIdx.x;      // 0..UMAX-1
  int b = blockIdx.y;      // 0..B-1
  int c = threadIdx.x;     // 0..C-1
  int d  = dur[(size_t)b * UMAX_ + u];
  int s0 = start[(size_t)b * UMAX_ + u];
  const float* pc = ec + ((size_t)b * C_ + c) * T_;
  const float* pp = ep + ((size_t)b * C_ + c) * T_;
  float sc = 0.f, sp = 0.f;
  for (int i = 0; i < d; ++i) { sc += pc[s0 + i]; sp += pp[s0 + i]; }
  float inv = (d > 0) ? 1.f / (float)d : 0.f;
  size_t o = ((size_t)b * C_ + c) * UMAX_ + u;
  u_c[o] = sc * inv;
  u_p[o] = sp * inv;
}

// ---------------- 3. speaker conditioning: cond[b][c] = g.cond_w + cond_b ----
__global__ __launch_bounds__(256) void cond_kernel(const float* __restrict__ g,
                                                   const float* __restrict__ cw,
                                                   const float* __restrict__ cb,
                                                   float* __restrict__ cond) {
  int b = blockIdx.x;
  int c = threadIdx.x;
  float s = 0.f;
  const float* gb = g + (size_t)b * GIN_;     // g is (B, GIN, 1)
  const float* wc = cw + (size_t)c * GIN_;    // cond_w is (C, GIN)
  for (int k = 0; k < GIN_; ++k) s += gb[k] * wc[k];
  cond[(size_t)b * C_ + c] = s + cb[c];
}

// ---------------- 4. build masked conv1 input --------------------------------
__global__ __launch_bounds__(256) void x0_kernel(const float* __restrict__ u_c,
                                                 const float* __restrict__ cond,
                                                 const float* __restrict__ mask,
                                                 float* __restrict__ x0m) {
  int idx = blockIdx.x * 256 + threadIdx.x;
  int total = B_ * C_ * UMAX_;
  if (idx >= total) return;
  int u = idx % UMAX_;
  int c = (idx / UMAX_) % C_;
  int b = idx / (C_ * UMAX_);
  x0m[idx] = (u_c[idx] + cond[(size_t)b * C_ + c]) * mask[(size_t)b * UMAX_ + u];
}

// ---------------- 4b. weight transpose: (F,C,3) -> (3,F,C) ------------------
__global__ __launch_bounds__(256) void wt_kernel(const float* __restrict__ W,
                                                 float* __restrict__ Wt) {
  int idx = blockIdx.x * 256 + threadIdx.x;
  if (idx >= F_ * C_ * 3) return;
  int tap = idx % 3;
  int c   = (idx / 3) % C_;
  int f   = idx / (3 * C_);
  Wt[((size_t)tap * F_ + f) * C_ + c] = W[idx];
}

// ---------------- 5. conv1d(k=3,pad=1) as WMMA f32 GEMM + bias + ReLU --------
// Block = 8 waves = 128(f) x 32(u) tile.  X slab (u0-1..u0+32, all 256 ch)
// staged in LDS once, reused by all waves/taps.  Each wave: 16(f) x 32(u),
// two accumulators share each A fragment.  Inner step: 1 global b64 (Wt) +
// 2 LDS b64 (X) + 2 v_wmma_f32_16x16x4_f32.
#define USLAB_ 34      // 32 + 2 halo rows
#define XPAD_  258     // 256 + 2 pad floats: stride 258 dwords -> spread banks
__global__ __launch_bounds__(256) void conv_wmma_kernel(const float* __restrict__ X,   // (B,C,UMAX) masked
                                                        const float* __restrict__ Wt,  // (3,F,C)
                                                        const float* __restrict__ bias,// (F)
                                                        float* __restrict__ Y) {       // (B,F,UMAX)
  __shared__ float xs[USLAB_][XPAD_];

  const int tid = threadIdx.x;                 // 0..255
  const int u0  = blockIdx.y * 32;
  const int b   = blockIdx.z;
  const float* Xb = X + (size_t)b * C_ * UMAX_;

  // cooperative stage: xs[u'][c] = X[c][u0-1+u'] (zero outside [0,UMAX))
  for (int idx = tid; idx < USLAB_ * C_; idx += 256) {
    int c  = idx / USLAB_;
    int up = idx % USLAB_;
    int us = u0 + up - 1;
    float v = (us >= 0 && us < UMAX_) ? Xb[(size_t)c * UMAX_ + us] : 0.f;
    xs[up][c] = v;
  }
  __syncthreads();

  const int wave = tid >> 5;                   // 0..7
  const int lane = tid & 31;
  const int half = lane >> 4;                  // 0 | 1
  const int l16  = lane & 15;
  const int kb   = half * 2;                   // K sub-pair {0,1} or {2,3}
  const int f0   = blockIdx.x * 128 + wave * 16;

  v8f acc0 = {0.f, 0.f, 0.f, 0.f, 0.f, 0.f, 0.f, 0.f};
  v8f acc1 = {0.f, 0.f, 0.f, 0.f, 0.f, 0.f, 0.f, 0.f};

  for (int tap = 0; tap < 3; ++tap) {
    const float* wrow = Wt + ((size_t)tap * F_ + (f0 + l16)) * C_;
    __builtin_prefetch(wrow, 0, 3);            // global_prefetch_b8 on the weight stream
    const int r0 = l16 + tap;                  // xs row for acc0 column n=l16
    for (int c0 = 0; c0 < C_; c0 += 4) {
      v2f a;                                   // A: W[f0+l16][c0+kb .. +1][tap], contiguous b64
      a.x = wrow[c0 + kb];
      a.y = wrow[c0 + kb + 1];
      v2f b0;                                  // B cols u0..u0+15
      b0.x = xs[r0][c0 + kb];
      b0.y = xs[r0][c0 + kb + 1];
      v2f b1;                                  // B cols u0+16..u0+31
      b1.x = xs[r0 + 16][c0 + kb];
      b1.y = xs[r0 + 16][c0 + kb + 1];
      acc0 = __builtin_amdgcn_wmma_f32_16x16x4_f32(false, a, false, b0, (short)0, acc0, false, false);
      acc1 = __builtin_amdgcn_wmma_f32_16x16x4_f32(false, a, false, b1, (short)0, acc1, false, false);
    }
  }

  // D layout: VGPR r: lanes0-15 -> M=r, N=l16 ; lanes16-31 -> M=r+8
  #pragma unroll
  for (int r = 0; r < 8; ++r) {
    int m = r + half * 8;
    float bv = bias[f0 + m];
    float v0 = fmaxf(acc0[r] + bv, 0.f);       // ReLU
    float v1 = fmaxf(acc1[r] + bv, 0.f);
    size_t o = ((size_t)b * F_ + (f0 + m)) * UMAX_ + u0 + l16;
    Y[o]      = v0;
    Y[o + 16] = v1;
  }
}

// ---------------- 6. LayerNorm over channels, output pre-masked -------------
__global__ __launch_bounds__(256) void ln_mask_kernel(const float* __restrict__ Y,
                                                      const float* __restrict__ gam,
                                                      const float* __restrict__ bet,
                                                      const float* __restrict__ mask,
                                                      float* __restrict__ Xo) {
  int bu = blockIdx.x;               // b*UMAX + u
  int b = bu / UMAX_;
  int u = bu % UMAX_;
  int c = threadIdx.x;
  float v = Y[((size_t)b * C_ + c) * UMAX_ + u];
  __shared__ float red[256];
  red[c] = v;
  __syncthreads();
  for (int s = 128; s > 0; s >>= 1) { if (c < s) red[c] += red[c + s]; __syncthreads(); }
  float mean = red[0] / (float)C_;
  __syncthreads();
  float d = v - mean;
  red[c] = d * d;
  __syncthreads();
  for (int s = 128; s > 0; s >>= 1) { if (c < s) red[c] += red[c + s]; __syncthreads(); }
  float var = red[0] / (float)C_;
  float o = d * rsqrtf(var + 1e-5f) * gam[c] + bet[c];
  Xo[((size_t)b * C_ + c) * UMAX_ + u] = o * mask[(size_t)b * UMAX_ + u];
}

// ---------------- 7. projection -> ceil -> integer durations ----------------
__global__ __launch_bounds__(256) void proj_kernel(const float* __restrict__ X2,   // masked
                                                   const float* __restrict__ pw,
                                                   const float* __restrict__ pb,
                                                   const float* __restrict__ mask,
                                                   int* __restrict__ dur2) {
  int bu = blockIdx.x;
  int b = bu / UMAX_;
  int u = bu % UMAX_;
  int c = threadIdx.x;
  __shared__ float red[256];
  red[c] = X2[((size_t)b * C_ + c) * UMAX_ + u] * pw[c];
  __syncthreads();
  for (int s = 128; s > 0; s >>= 1) { if (c < s) red[c] += red[c + s]; __syncthreads(); }
  if (c == 0) {
    float m = mask[(size_t)b * UMAX_ + u];
    float dp = (red[0] + pb[0]) * m;    // X2 already masked
    dur2[(size_t)b * UMAX_ + u] = (int)ceilf(dp * m);
  }
}

// ---------------- 8. per-row cumsum of predicted durations ------------------
__global__ __launch_bounds__(16) void scan_kernel(const int* __restrict__ dur2,
                                                  int* __restrict__ prev2,
                                                  int* __restrict__ csum2) {
  int b = threadIdx.x;
  if (b >= B_) return;
  int acc = 0;
  for (int u = 0; u < UMAX_; ++u) {
    prev2[(size_t)b * UMAX_ + u] = acc;
    acc += dur2[(size_t)b * UMAX_ + u];
    csum2[(size_t)b * UMAX_ + u] = acc;
  }
}

// ---------------- 9. frame -> unit mapping (exact interval sum) -------------
__global__ __launch_bounds__(256) void um2_kernel(const int* __restrict__ prev2,
                                                  const int* __restrict__ csum2,
                                                  int* __restrict__ um2, int Tnew) {
  int idx = blockIdx.x * 256 + threadIdx.x;
  if (idx >= B_ * Tnew) return;
  int b = idx / Tnew;
  int t = idx % Tnew;
  int s = 0;
  const int* pv = prev2 + (size_t)b * UMAX_;
  const int* cv = csum2 + (size_t)b * UMAX_;
  for (int u = 0; u < UMAX_; ++u)
    s += (t >= pv[u] && t < cv[u]) ? (u + 1) : 0;
  if (s > UMAX_) s = UMAX_;
  if (s < 0) s = 0;
  um2[(size_t)b * Tnew + t] = s;
}

// ---------------- 10. expand units + linear-interp mel -> output ------------
__global__ __launch_bounds__(256) void out_kernel(const float* __restrict__ u_c,
                                                  const float* __restrict__ u_p,
                                                  const float* __restrict__ mel,
                                                  const int* __restrict__ um2,
                                                  float* __restrict__ out, int Tnew) {
  int idx = blockIdx.x * 256 + threadIdx.x;
  int total = B_ * OUTC_ * Tnew;
  if (idx >= total) return;
  int t = idx % Tnew;
  int ch = (idx / Tnew) % OUTC_;
  int b = idx / (OUTC_ * Tnew);
  float v;
  if (ch < 2 * C_) {
    int m2 = um2[(size_t)b * Tnew + t];
    const float* src = (ch < C_) ? u_c : u_p;
    int c = (ch < C_) ? ch : (ch - C_);
    v = (m2 > 0) ? src[((size_t)b * C_ + c) * UMAX_ + (m2 - 1)] : 0.f;
  } else {
    int c = ch - 2 * C_;
    float sp = ((float)t + 0.5f) * ((float)T_ / (float)Tnew) - 0.5f;
    sp = fminf(fmaxf(sp, 0.f), (float)(T_ - 1));
    int i0 = (int)floorf(sp);
    int i1 = (i0 + 1 < T_) ? i0 + 1 : T_ - 1;
    float w = sp - (float)i0;
    const float* mb = mel + ((size_t)b * MELC_ + c) * T_;
    v = mb[i0] * (1.f - w) + mb[i1] * w;
  }
  out[idx] = v;
}

extern "C" void kernel_launch(void* const* d_in, const int* in_sizes, int n_in,
                              void* d_out, int out_size, void* d_ws, size_t ws_size,
                              hipStream_t stream) {
  const float* emb_c = (const float*)d_in[0];
  const float* emb_p = (const float*)d_in[1];
  const float* mel   = (const float*)d_in[2];
  const float* g     = (const float*)d_in[3];
  const float* w1    = (const float*)d_in[4];
  const float* b1    = (const float*)d_in[5];
  const float* ln1g  = (const float*)d_in[6];
  const float* ln1b  = (const float*)d_in[7];
  const float* w2    = (const float*)d_in[8];
  const float* b2    = (const float*)d_in[9];
  const float* ln2g  = (const float*)d_in[10];
  const float* ln2b  = (const float*)d_in[11];
  const float* pw    = (const float*)d_in[12];
  const float* pb    = (const float*)d_in[13];
  const float* cw    = (const float*)d_in[14];
  const float* cb    = (const float*)d_in[15];
  const int*   ph    = (const int*)d_in[16];

  int Tnew = out_size / (B_ * OUTC_);   // output time length, from harness

  // ---- workspace layout ----
  char* ws = (char*)d_ws;
  const size_t big = (size_t)B_ * C_ * UMAX_ * sizeof(float);   // 16.8 MB each
  float* u_c  = (float*)ws;               ws += big;
  float* u_p  = (float*)ws;               ws += big;
  float* xA   = (float*)ws;               ws += big;  // conv inputs / LN outputs
  float* yB   = (float*)ws;               ws += big;  // raw conv outputs
  const size_t wsz = (size_t)F_ * C_ * 3 * sizeof(float);       // 786 KB each
  float* w1t  = (float*)ws;               ws += wsz;
  float* w2t  = (float*)ws;               ws += wsz;
  const size_t row = (size_t)B_ * UMAX_ * sizeof(int);
  float* mask  = (float*)ws;              ws += row;
  int*   dur   = (int*)ws;                ws += row;
  int*   start = (int*)ws;                ws += row;
  int*   dur2  = (int*)ws;                ws += row;
  int*   prev2 = (int*)ws;                ws += row;
  int*   csum2 = (int*)ws;                ws += row;
  float* condv = (float*)ws;              ws += (size_t)B_ * C_ * sizeof(float);
  int*   um2   = (int*)ws;                ws += (size_t)B_ * (size_t)(Tnew > 0 ? Tnew : 1) * sizeof(int);

  // 1. run-length encode phonemes
  dedup_kernel<<<1, 16, 0, stream>>>(ph, dur, start, mask);
  // 2. pool content/pitch embeddings into units
  pool_kernel<<<dim3(UMAX_, B_), 256, 0, stream>>>(emb_c, emb_p, dur, start, u_c, u_p);
  // 3. speaker conditioning
  cond_kernel<<<B_, 256, 0, stream>>>(g, cw, cb, condv);
  // 4. masked predictor input + weight transposes
  {
    int total = B_ * C_ * UMAX_;
    x0_kernel<<<(total + 255) / 256, 256, 0, stream>>>(u_c, condv, mask, xA);
    int wtot = F_ * C_ * 3;
    wt_kernel<<<(wtot + 255) / 256, 256, 0, stream>>>(w1, w1t);
    wt_kernel<<<(wtot + 255) / 256, 256, 0, stream>>>(w2, w2t);
  }
  dim3 cgrid(F_ / 128, UMAX_ / 32, B_);
  // 5-6. conv1 (WMMA, LDS-staged) + ReLU, then LayerNorm (+ mask)
  conv_wmma_kernel<<<cgrid, 256, 0, stream>>>(xA, w1t, b1, yB);
  ln_mask_kernel<<<B_ * UMAX_, 256, 0, stream>>>(yB, ln1g, ln1b, mask, xA);
  // 7-8. conv2 (WMMA, LDS-staged) + ReLU, then LayerNorm (+ mask)
  conv_wmma_kernel<<<cgrid, 256, 0, stream>>>(xA, w2t, b2, yB);
  ln_mask_kernel<<<B_ * UMAX_, 256, 0, stream>>>(yB, ln2g, ln2b, mask, xA);
  // 9. projection -> ceil -> integer durations, then per-row cumsum
  proj_kernel<<<B_ * UMAX_, 256, 0, stream>>>(xA, pw, pb, mask, dur2);
  scan_kernel<<<1, 16, 0, stream>>>(dur2, prev2, csum2);
  if (Tnew > 0) {
    // 10. frame->unit mapping for the new timeline
    um2_kernel<<<(B_ * Tnew + 255) / 256, 256, 0, stream>>>(prev2, csum2, um2, Tnew);
    // 11. expand + mel interpolation -> output
    int total = B_ * OUTC_ * Tnew;
    out_kernel<<<(total + 255) / 256, 256, 0, stream>>>(u_c, u_p, mel, um2, (float*)d_out, Tnew);
  }
}